// GIN_31911607009304
// MI455X (gfx1250) — compile-verified
//
#include <hip/hip_runtime.h>
#include <hip/hip_bf16.h>

#define N_NODES   50000
#define N_EDGES   800000
#define HIDDEN    128
#define N_GRAPHS  256
#define N_CLASSES 10
#define LDPAD     132   // 128 + 4 floats: stride%64 = 4 -> conflict-free LDS column reads
#define MTILES    4     // 16-row M-tiles per block (M = 64 rows/block)
#define NT16      ((N_NODES + 15) / 16)   // 3125 total 16-row tiles (exact)

typedef __attribute__((ext_vector_type(2))) float v2f;
typedef __attribute__((ext_vector_type(8))) float v8f;

// ---------------------------------------------------------------------------
// Edge scatter-add: agg[dst] += x[src].  One thread = (edge, 4 channels).
// x (25.6MB) + agg (25.6MB) are L2-resident (192MB L2); f32 atomics resolve
// in L2, so this phase is L2-atomic bound, not HBM bound.
// ---------------------------------------------------------------------------
__global__ __launch_bounds__(256) void gin_scatter_add(
    const float* __restrict__ x,
    const int*   __restrict__ src,
    const int*   __restrict__ dst,
    float*       __restrict__ agg) {
  int idx = blockIdx.x * blockDim.x + threadIdx.x;
  int e = idx >> 5;
  if (e >= N_EDGES) return;
  int c = (idx & 31) << 2;
  int s = src[e];
  int d = dst[e];
  const float4 v = *reinterpret_cast<const float4*>(x + (size_t)s * HIDDEN + c);
  float* p = agg + (size_t)d * HIDDEN + c;
  atomicAdd(p + 0, v.x);
  atomicAdd(p + 1, v.y);
  atomicAdd(p + 2, v.z);
  atomicAdd(p + 3, v.w);
}

// ---------------------------------------------------------------------------
// Fused GIN MLP:  out = [relu]( relu((x+agg) @ W1 + b1) @ W2 + b2 )
// Block = 256 threads = 8 wave32s, owns 64 node rows (4 M-tiles).
// Wave w owns output columns [16w, 16w+16) for all 4 M-tiles: each B
// fragment (weight pair) is loaded once and reused by 4 independent
// v_wmma_f32_16x16x4_f32 accumulator chains -> 4x less L2 weight traffic,
// 4x WMMA density, 4-deep ILP for the matrix pipe.
// ---------------------------------------------------------------------------
template <bool RELU_OUT>
__global__ __launch_bounds__(256) void gin_mlp(
    const float* __restrict__ xin,
    const float* __restrict__ agg,
    const float* __restrict__ W1, const float* __restrict__ b1,
    const float* __restrict__ W2, const float* __restrict__ b2,
    float*       __restrict__ out) {
  __shared__ float sH[16 * MTILES * LDPAD];   // input tiles  h = x + agg
  __shared__ float sT[16 * MTILES * LDPAD];   // mid tiles    t = relu(h@W1+b1)

  const int t0  = blockIdx.x * MTILES;        // first 16-row tile index
  const int tid = threadIdx.x;

  // ---- phase 0: cooperative load, h = x + agg (2 float4 per thread/tile) --
  {
    const int row = tid >> 4;                 // 0..15
    const int col = (tid & 15) << 3;          // 0,8,...,120
#pragma unroll
    for (int i = 0; i < MTILES; ++i) {
      float* dsp = sH + (i * 16 + row) * LDPAD + col;
      if (t0 + i < NT16) {
        const size_t g = (size_t)((t0 + i) * 16 + row) * HIDDEN + col;
        float4 a0 = *reinterpret_cast<const float4*>(xin + g);
        float4 a1 = *reinterpret_cast<const float4*>(xin + g + 4);
        float4 s0 = *reinterpret_cast<const float4*>(agg + g);
        float4 s1 = *reinterpret_cast<const float4*>(agg + g + 4);
        dsp[0] = a0.x + s0.x; dsp[1] = a0.y + s0.y;
        dsp[2] = a0.z + s0.z; dsp[3] = a0.w + s0.w;
        dsp[4] = a1.x + s1.x; dsp[5] = a1.y + s1.y;
        dsp[6] = a1.z + s1.z; dsp[7] = a1.w + s1.w;
      } else {                                 // zero-fill dead tile
        dsp[0] = 0.f; dsp[1] = 0.f; dsp[2] = 0.f; dsp[3] = 0.f;
        dsp[4] = 0.f; dsp[5] = 0.f; dsp[6] = 0.f; dsp[7] = 0.f;
      }
    }
  }
  __syncthreads();

  const int lane = tid & 31;
  const int wave = tid >> 5;
  const int n0   = wave * 16;       // output column tile base
  const int half = lane >> 4;       // 0: lanes 0-15, 1: lanes 16-31
  const int l    = lane & 15;

  // ---- phase 1: t = relu(h @ W1 + b1) : 128 WMMAs, 4 indep chains ----
  v8f acc0 = {}, acc1 = {}, acc2 = {}, acc3 = {};
#pragma unroll
  for (int k = 0; k < HIDDEN; k += 4) {
    const int ka = k + 2 * half;               // K = {ka, ka+1}
    v2f b;
    b.x = W1[(size_t)ka * HIDDEN + n0 + l];    // B: N = l, loaded once
    b.y = W1[(size_t)(ka + 1) * HIDDEN + n0 + l];
    v2f a;
    a.x = sH[(0 * 16 + l) * LDPAD + ka]; a.y = sH[(0 * 16 + l) * LDPAD + ka + 1];
    acc0 = __builtin_amdgcn_wmma_f32_16x16x4_f32(false, a, false, b, (short)0, acc0, false, false);
    a.x = sH[(1 * 16 + l) * LDPAD + ka]; a.y = sH[(1 * 16 + l) * LDPAD + ka + 1];
    acc1 = __builtin_amdgcn_wmma_f32_16x16x4_f32(false, a, false, b, (short)0, acc1, false, false);
    a.x = sH[(2 * 16 + l) * LDPAD + ka]; a.y = sH[(2 * 16 + l) * LDPAD + ka + 1];
    acc2 = __builtin_amdgcn_wmma_f32_16x16x4_f32(false, a, false, b, (short)0, acc2, false, false);
    a.x = sH[(3 * 16 + l) * LDPAD + ka]; a.y = sH[(3 * 16 + l) * LDPAD + ka + 1];
    acc3 = __builtin_amdgcn_wmma_f32_16x16x4_f32(false, a, false, b, (short)0, acc3, false, false);
  }
  {
    const float bias = b1[n0 + l];
#pragma unroll
    for (int r = 0; r < 8; ++r) {              // D: M = r + 8*half, N = l
      float v0 = acc0[r] + bias; v0 = v0 > 0.f ? v0 : 0.f;
      float v1 = acc1[r] + bias; v1 = v1 > 0.f ? v1 : 0.f;
      float v2 = acc2[r] + bias; v2 = v2 > 0.f ? v2 : 0.f;
      float v3 = acc3[r] + bias; v3 = v3 > 0.f ? v3 : 0.f;
      const int rr = r + 8 * half;
      sT[(0 * 16 + rr) * LDPAD + n0 + l] = v0;
      sT[(1 * 16 + rr) * LDPAD + n0 + l] = v1;
      sT[(2 * 16 + rr) * LDPAD + n0 + l] = v2;
      sT[(3 * 16 + rr) * LDPAD + n0 + l] = v3;
    }
  }
  __syncthreads();

  // ---- phase 2: out = t @ W2 + b2 (optionally relu'd) ----
  acc0 = (v8f){}; acc1 = (v8f){}; acc2 = (v8f){}; acc3 = (v8f){};
#pragma unroll
  for (int k = 0; k < HIDDEN; k += 4) {
    const int ka = k + 2 * half;
    v2f b;
    b.x = W2[(size_t)ka * HIDDEN + n0 + l];
    b.y = W2[(size_t)(ka + 1) * HIDDEN + n0 + l];
    v2f a;
    a.x = sT[(0 * 16 + l) * LDPAD + ka]; a.y = sT[(0 * 16 + l) * LDPAD + ka + 1];
    acc0 = __builtin_amdgcn_wmma_f32_16x16x4_f32(false, a, false, b, (short)0, acc0, false, false);
    a.x = sT[(1 * 16 + l) * LDPAD + ka]; a.y = sT[(1 * 16 + l) * LDPAD + ka + 1];
    acc1 = __builtin_amdgcn_wmma_f32_16x16x4_f32(false, a, false, b, (short)0, acc1, false, false);
    a.x = sT[(2 * 16 + l) * LDPAD + ka]; a.y = sT[(2 * 16 + l) * LDPAD + ka + 1];
    acc2 = __builtin_amdgcn_wmma_f32_16x16x4_f32(false, a, false, b, (short)0, acc2, false, false);
    a.x = sT[(3 * 16 + l) * LDPAD + ka]; a.y = sT[(3 * 16 + l) * LDPAD + ka + 1];
    acc3 = __builtin_amdgcn_wmma_f32_16x16x4_f32(false, a, false, b, (short)0, acc3, false, false);
  }
  {
    const float bias = b2[n0 + l];
#pragma unroll
    for (int i = 0; i < MTILES; ++i) {
      if (t0 + i >= NT16) break;               // uniform branch: EXEC untouched
      const v8f& acc = (i == 0) ? acc0 : (i == 1) ? acc1 : (i == 2) ? acc2 : acc3;
#pragma unroll
      for (int r = 0; r < 8; ++r) {
        float v = acc[r] + bias;
        if (RELU_OUT) v = v > 0.f ? v : 0.f;
        out[(size_t)((t0 + i) * 16 + r + 8 * half) * HIDDEN + n0 + l] = v;
      }
    }
  }
}

// ---------------------------------------------------------------------------
// Global mean-pool numerator + counts (atomic segment sum over sorted batch).
// ---------------------------------------------------------------------------
__global__ __launch_bounds__(256) void gin_pool(
    const float* __restrict__ h,
    const int*   __restrict__ batch,
    float*       __restrict__ pooled,
    float*       __restrict__ cnt) {
  int idx = blockIdx.x * blockDim.x + threadIdx.x;
  int n = idx >> 5;
  if (n >= N_NODES) return;
  int c = (idx & 31) << 2;
  int g = batch[n];
  const float4 v = *reinterpret_cast<const float4*>(h + (size_t)n * HIDDEN + c);
  float* p = pooled + (size_t)g * HIDDEN + c;
  atomicAdd(p + 0, v.x);
  atomicAdd(p + 1, v.y);
  atomicAdd(p + 2, v.z);
  atomicAdd(p + 3, v.w);
  if (c == 0) atomicAdd(cnt + g, 1.0f);
}

// ---------------------------------------------------------------------------
// Head: out = (pooled/cnt) @ Wl + bl.   2560 outputs -> trivial VALU kernel.
// ---------------------------------------------------------------------------
__global__ __launch_bounds__(256) void gin_head(
    const float* __restrict__ pooled,
    const float* __restrict__ cnt,
    const float* __restrict__ Wl,
    const float* __restrict__ bl,
    float*       __restrict__ out) {
  int idx = blockIdx.x * blockDim.x + threadIdx.x;
  if (idx >= N_GRAPHS * N_CLASSES) return;
  int g = idx / N_CLASSES;
  int c = idx % N_CLASSES;
  float inv = 1.0f / fmaxf(cnt[g], 1.0f);
  float s = 0.0f;
#pragma unroll 8
  for (int k = 0; k < HIDDEN; ++k)
    s += pooled[(size_t)g * HIDDEN + k] * Wl[(size_t)k * N_CLASSES + c];
  out[idx] = s * inv + bl[c];
}

extern "C" void kernel_launch(void* const* d_in, const int* in_sizes, int n_in,
                              void* d_out, int out_size, void* d_ws, size_t ws_size,
                              hipStream_t stream) {
  const float* x     = (const float*)d_in[0];
  const int*   edge  = (const int*)d_in[1];   // (2, N_EDGES) flat
  const int*   batch = (const int*)d_in[2];
  const float* W1    = (const float*)d_in[3];
  const float* b1    = (const float*)d_in[4];
  const float* W2    = (const float*)d_in[5];
  const float* b2    = (const float*)d_in[6];
  const float* Wl    = (const float*)d_in[7];
  const float* bl    = (const float*)d_in[8];
  float*       out   = (float*)d_out;

  const int* src = edge;
  const int* dst = edge + N_EDGES;

  const size_t feat = (size_t)N_NODES * HIDDEN;
  float* agg    = (float*)d_ws;                 // 25.6 MB
  float* hA     = agg + feat;                   // 25.6 MB
  float* hB     = hA + feat;                    // 25.6 MB
  float* pooled = hB + feat;                    // 128 KB
  float* cnt    = pooled + (size_t)N_GRAPHS * HIDDEN;  // 1 KB

  const dim3 blk(256);
  const dim3 sgrd((N_EDGES * 32 + 255) / 256);      // scatter grid
  const dim3 mgrd((NT16 + MTILES - 1) / MTILES);    // 782 blocks of 64 rows
  const dim3 pgrd((N_NODES * 32 + 255) / 256);
  const dim3 hgrd((N_GRAPHS * N_CLASSES + 255) / 256);

  // ---- layer 1 (outer relu) ----
  hipMemsetAsync(agg, 0, feat * sizeof(float), stream);
  gin_scatter_add<<<sgrd, blk, 0, stream>>>(x, src, dst, agg);
  gin_mlp<true><<<mgrd, blk, 0, stream>>>(x, agg, W1, b1, W2, b2, hA);

  // ---- layer 2 (outer relu) ----
  hipMemsetAsync(agg, 0, feat * sizeof(float), stream);
  gin_scatter_add<<<sgrd, blk, 0, stream>>>(hA, src, dst, agg);
  gin_mlp<true><<<mgrd, blk, 0, stream>>>(hA, agg, W1, b1, W2, b2, hB);

  // ---- layer 3 (no outer relu) ----
  hipMemsetAsync(agg, 0, feat * sizeof(float), stream);
  gin_scatter_add<<<sgrd, blk, 0, stream>>>(hB, src, dst, agg);
  gin_mlp<false><<<mgrd, blk, 0, stream>>>(hB, agg, W1, b1, W2, b2, hA);

  // ---- global mean pool + linear head ----
  hipMemsetAsync(pooled, 0,
                 ((size_t)N_GRAPHS * HIDDEN + N_GRAPHS) * sizeof(float), stream);
  gin_pool<<<pgrd, blk, 0, stream>>>(hA, batch, pooled, cnt);
  gin_head<<<hgrd, blk, 0, stream>>>(pooled, cnt, Wl, bl, out);
}